// TransformerConv_8022998909562
// MI455X (gfx1250) — compile-verified
//
#include <hip/hip_runtime.h>
#include <math.h>

typedef __attribute__((ext_vector_type(2))) float v2f;
typedef __attribute__((ext_vector_type(8))) float v8f;

#define IN_DIM   128
#define HID_DIM  128
#define LDS_PITCH 132   // 128 + 4 padding -> conflict-free strided fragment reads

// ---------------------------------------------------------------------------
// Float atomic max via the signed/unsigned int trick (bitwise order-preserving)
// ---------------------------------------------------------------------------
__device__ __forceinline__ void atomicMaxF32(float* addr, float val) {
    if (val >= 0.0f)
        atomicMax((int*)addr, __float_as_int(val));
    else
        atomicMin((unsigned int*)addr, __float_as_uint(val));
}

// ---------------------------------------------------------------------------
// Workspace init: m = -inf, denom = 0, agg = 0
// ---------------------------------------------------------------------------
__global__ __launch_bounds__(256)
void init_ws_kernel(float* __restrict__ m, float* __restrict__ denom,
                    float* __restrict__ agg, int N) {
    int i = blockIdx.x * blockDim.x + threadIdx.x;
    if (i < N * 4) { m[i] = -INFINITY; denom[i] = 0.0f; }
    if (i < N * HID_DIM) agg[i] = 0.0f;
}

// ---------------------------------------------------------------------------
// Fused Q/K/V/Skip projection GEMM using V_WMMA_F32_16X16X4_F32.
// grid = (ceil(N/32), 4); block = 256 (8 waves).
// blockIdx.y selects {Wq,Wk,Wv,Ws}; each wave owns one 16-col tile of HID=128
// and TWO 16-row tiles (M=32 per block), sharing each B fragment across two
// WMMAs (2x matrix-op density per weight fetch).
// A 32x128 feat tile is staged in LDS and reused by all 8 waves.
// ---------------------------------------------------------------------------
__global__ __launch_bounds__(256)
void qkvs_gemm_kernel(const float* __restrict__ feat,
                      const float* __restrict__ Wq, const float* __restrict__ bq,
                      const float* __restrict__ Wk, const float* __restrict__ bk,
                      const float* __restrict__ Wv, const float* __restrict__ bv,
                      const float* __restrict__ Ws, const float* __restrict__ bs,
                      float* __restrict__ q, float* __restrict__ k,
                      float* __restrict__ v, float* __restrict__ sk, int N) {
    __shared__ float lds[32 * LDS_PITCH];

    const float* W; const float* B; float* O;
    switch (blockIdx.y) {
        case 0:  W = Wq; B = bq; O = q;  break;
        case 1:  W = Wk; B = bk; O = k;  break;
        case 2:  W = Wv; B = bv; O = v;  break;
        default: W = Ws; B = bs; O = sk; break;
    }

    const int row0 = blockIdx.x * 32;
    const int t    = threadIdx.x;

    // Stage 32x128 feat tile: 256 threads x 16 floats each (four float4s).
    {
        const int r  = t >> 3;           // 0..31 row within tile
        const int cg = (t & 7) * 16;     // column group (0,16,...,112)
        float4 x0 = make_float4(0.f, 0.f, 0.f, 0.f);
        float4 x1 = x0, x2 = x0, x3 = x0;
        if (row0 + r < N) {
            const float4* s4 = (const float4*)(feat + (size_t)(row0 + r) * IN_DIM + cg);
            x0 = s4[0]; x1 = s4[1]; x2 = s4[2]; x3 = s4[3];
        }
        float* d = lds + r * LDS_PITCH + cg;
        *(float4*)(d)      = x0;
        *(float4*)(d + 4)  = x1;
        *(float4*)(d + 8)  = x2;
        *(float4*)(d + 12) = x3;
    }
    __syncthreads();

    const int wave  = t >> 5;           // 0..7 -> output column tile
    const int lane  = t & 31;
    const int n0    = wave * 16;
    const int mrow  = lane & 15;        // A-matrix row this lane feeds
    const int ncol  = lane & 15;        // B/D column this lane feeds
    const int khalf = (lane >> 4) * 2;  // lanes 0-15: K=0,1 ; lanes 16-31: K=2,3

    v8f acc0 = {};
    v8f acc1 = {};
    const float* lA0 = lds + mrow * LDS_PITCH + khalf;          // rows 0..15
    const float* lA1 = lds + (mrow + 16) * LDS_PITCH + khalf;   // rows 16..31

    #pragma unroll 4
    for (int kb = 0; kb < IN_DIM; kb += 4) {
        v2f b;                                    // B fragment (4x16 f32), shared
        b.x = W[(size_t)(kb + khalf)     * HID_DIM + n0 + ncol];
        b.y = W[(size_t)(kb + khalf + 1) * HID_DIM + n0 + ncol];
        v2f a0;                                   // A fragment, row tile 0
        a0.x = lA0[kb];
        a0.y = lA0[kb + 1];
        v2f a1;                                   // A fragment, row tile 1
        a1.x = lA1[kb];
        a1.y = lA1[kb + 1];
        acc0 = __builtin_amdgcn_wmma_f32_16x16x4_f32(
            false, a0, false, b, (short)0, acc0, false, false);
        acc1 = __builtin_amdgcn_wmma_f32_16x16x4_f32(
            false, a1, false, b, (short)0, acc1, false, false);
    }

    const float bias  = B[n0 + ncol];
    const int   rbase = (lane >> 4) << 3;   // upper-lane half -> rows +8

    // Row tile 0 epilogue: single range test -> 8 immediate-offset stores.
    {
        const int m0 = row0 + rbase;
        float* Ob = O + (size_t)m0 * HID_DIM + n0 + ncol;
        if (m0 + 8 <= N) {
            #pragma unroll
            for (int r = 0; r < 8; ++r) Ob[(size_t)r * HID_DIM] = acc0[r] + bias;
        } else {
            #pragma unroll
            for (int r = 0; r < 8; ++r)
                if (m0 + r < N) Ob[(size_t)r * HID_DIM] = acc0[r] + bias;
        }
    }
    // Row tile 1 epilogue.
    {
        const int m1 = row0 + 16 + rbase;
        float* Ob = O + (size_t)m1 * HID_DIM + n0 + ncol;
        if (m1 + 8 <= N) {
            #pragma unroll
            for (int r = 0; r < 8; ++r) Ob[(size_t)r * HID_DIM] = acc1[r] + bias;
        } else {
            #pragma unroll
            for (int r = 0; r < 8; ++r)
                if (m1 + r < N) Ob[(size_t)r * HID_DIM] = acc1[r] + bias;
        }
    }
}

// ---------------------------------------------------------------------------
// Edge attention logits + segment max. One wave per edge; lane owns 4 dims.
// Head h occupies lanes [8h, 8h+8): 8 lanes x 4 dims = D=32.
// ---------------------------------------------------------------------------
__global__ __launch_bounds__(256)
void edge_logits_kernel(const float* __restrict__ q, const float* __restrict__ kk,
                        const int* __restrict__ src, const int* __restrict__ dst,
                        float* __restrict__ a, float* __restrict__ m, int E) {
    const int wid  = (blockIdx.x * blockDim.x + threadIdx.x) >> 5;
    const int lane = threadIdx.x & 31;
    if (wid >= E) return;

    const int sN = src[wid];
    const int dN = dst[wid];
    const float4 q4 = *(const float4*)(q  + (size_t)sN * HID_DIM + lane * 4);
    const float4 k4 = *(const float4*)(kk + (size_t)dN * HID_DIM + lane * 4);
    float p = q4.x * k4.x + q4.y * k4.y + q4.z * k4.z + q4.w * k4.w;

    // reduce within each 8-lane head group
    p += __shfl_xor(p, 1, 32);
    p += __shfl_xor(p, 2, 32);
    p += __shfl_xor(p, 4, 32);

    if ((lane & 7) == 0) {
        const int h = lane >> 3;
        const float logit = p * 0.17677669529663687f;   // 1/sqrt(D=32)
        a[(size_t)wid * 4 + h] = logit;
        atomicMaxF32(m + (size_t)dN * 4 + h, logit);
    }
}

// ---------------------------------------------------------------------------
// ex = exp(a - m[dst]) in place; denom[dst] += ex.  One thread per (edge,head).
// ---------------------------------------------------------------------------
__global__ __launch_bounds__(256)
void edge_exp_kernel(float* __restrict__ a, const int* __restrict__ dst,
                     const float* __restrict__ m, float* __restrict__ denom, int E) {
    const int i = blockIdx.x * blockDim.x + threadIdx.x;
    if (i >= E * 4) return;
    const int e = i >> 2;
    const int h = i & 3;
    const int dN = dst[e];
    const float ex = __expf(a[i] - m[(size_t)dN * 4 + h]);
    a[i] = ex;
    atomicAdd(denom + (size_t)dN * 4 + h, ex);
}

// ---------------------------------------------------------------------------
// Weighted message aggregation: agg[dst] += v[src] * (ex / denom[dst]).
// One wave per edge; lane owns a float4 slice.
// ---------------------------------------------------------------------------
__global__ __launch_bounds__(256)
void edge_aggregate_kernel(const float* __restrict__ v,
                           const int* __restrict__ src, const int* __restrict__ dst,
                           const float* __restrict__ ex, const float* __restrict__ denom,
                           float* __restrict__ agg, int E) {
    const int wid  = (blockIdx.x * blockDim.x + threadIdx.x) >> 5;
    const int lane = threadIdx.x & 31;
    if (wid >= E) return;

    const int sN = src[wid];
    const int dN = dst[wid];
    const int h  = lane >> 3;
    const float dn = denom[(size_t)dN * 4 + h];
    const float w  = ex[(size_t)wid * 4 + h] / (dn > 0.0f ? dn : 1.0f);

    const float4 v4 = *(const float4*)(v + (size_t)sN * HID_DIM + lane * 4);
    float* out = agg + (size_t)dN * HID_DIM + lane * 4;
    atomicAdd(out + 0, v4.x * w);
    atomicAdd(out + 1, v4.y * w);
    atomicAdd(out + 2, v4.z * w);
    atomicAdd(out + 3, v4.w * w);
}

// ---------------------------------------------------------------------------
// Gate GEMV + sigmoid mix + LayerNorm (biased var, eps=1e-5) + PReLU.
// One wave per node; lane owns a float4 slice of the 128-wide row.
// ---------------------------------------------------------------------------
__global__ __launch_bounds__(256)
void gate_ln_kernel(const float* __restrict__ skip, const float* __restrict__ agg,
                    const float* __restrict__ Wg, const float* __restrict__ bg,
                    const float* __restrict__ gamma, const float* __restrict__ beta,
                    const float* __restrict__ alpha, float* __restrict__ out, int N) {
    const int wid  = (blockIdx.x * blockDim.x + threadIdx.x) >> 5;
    const int lane = threadIdx.x & 31;
    if (wid >= N) return;

    const size_t base = (size_t)wid * HID_DIM + lane * 4;
    const float4 s4 = *(const float4*)(skip + base);
    const float4 a4 = *(const float4*)(agg + base);
    const float4 wS = *(const float4*)(Wg + lane * 4);
    const float4 wA = *(const float4*)(Wg + 128 + lane * 4);
    const float4 wD = *(const float4*)(Wg + 256 + lane * 4);

    float gp = s4.x * wS.x + s4.y * wS.y + s4.z * wS.z + s4.w * wS.w
             + a4.x * wA.x + a4.y * wA.y + a4.z * wA.z + a4.w * wA.w
             + (s4.x - a4.x) * wD.x + (s4.y - a4.y) * wD.y
             + (s4.z - a4.z) * wD.z + (s4.w - a4.w) * wD.w;
    #pragma unroll
    for (int off = 1; off < 32; off <<= 1) gp += __shfl_xor(gp, off, 32);
    const float g = 1.0f / (1.0f + __expf(-(gp + bg[0])));

    float4 r;
    r.x = g * s4.x + (1.0f - g) * a4.x;
    r.y = g * s4.y + (1.0f - g) * a4.y;
    r.z = g * s4.z + (1.0f - g) * a4.z;
    r.w = g * s4.w + (1.0f - g) * a4.w;

    float sum = r.x + r.y + r.z + r.w;
    float sq  = r.x * r.x + r.y * r.y + r.z * r.z + r.w * r.w;
    #pragma unroll
    for (int off = 1; off < 32; off <<= 1) {
        sum += __shfl_xor(sum, off, 32);
        sq  += __shfl_xor(sq,  off, 32);
    }
    const float mu  = sum * (1.0f / HID_DIM);
    const float var = sq * (1.0f / HID_DIM) - mu * mu;
    const float rs  = rsqrtf(var + 1e-5f);

    const float4 gm = *(const float4*)(gamma + lane * 4);
    const float4 bt = *(const float4*)(beta + lane * 4);
    const float  al = alpha[0];

    float4 y;
    y.x = (r.x - mu) * rs * gm.x + bt.x;  y.x = (y.x >= 0.f) ? y.x : al * y.x;
    y.y = (r.y - mu) * rs * gm.y + bt.y;  y.y = (y.y >= 0.f) ? y.y : al * y.y;
    y.z = (r.z - mu) * rs * gm.z + bt.z;  y.z = (y.z >= 0.f) ? y.z : al * y.z;
    y.w = (r.w - mu) * rs * gm.w + bt.w;  y.w = (y.w >= 0.f) ? y.w : al * y.w;

    *(float4*)(out + base) = y;
}

// ---------------------------------------------------------------------------
// Host-side launch
// ---------------------------------------------------------------------------
extern "C" void kernel_launch(void* const* d_in, const int* in_sizes, int n_in,
                              void* d_out, int out_size, void* d_ws, size_t ws_size,
                              hipStream_t stream) {
    const float* feat  = (const float*)d_in[0];
    const int*   src   = (const int*)  d_in[1];
    const int*   dst   = (const int*)  d_in[2];
    const float* Wq    = (const float*)d_in[3];
    const float* bq    = (const float*)d_in[4];
    const float* Wk    = (const float*)d_in[5];
    const float* bk    = (const float*)d_in[6];
    const float* Wv    = (const float*)d_in[7];
    const float* bv    = (const float*)d_in[8];
    const float* Ws    = (const float*)d_in[9];
    const float* bs    = (const float*)d_in[10];
    const float* Wg    = (const float*)d_in[11];
    const float* bg    = (const float*)d_in[12];
    const float* gamma = (const float*)d_in[13];
    const float* beta  = (const float*)d_in[14];
    const float* alpha = (const float*)d_in[15];

    const int N = in_sizes[0] / IN_DIM;   // 50000
    const int E = in_sizes[1];            // 800000

    // Workspace layout (floats)
    float* ws    = (float*)d_ws;
    float* q     = ws;                       // N*128
    float* k     = q     + (size_t)N * 128;  // N*128
    float* v     = k     + (size_t)N * 128;  // N*128
    float* sk    = v     + (size_t)N * 128;  // N*128
    float* agg   = sk    + (size_t)N * 128;  // N*128
    float* m     = agg   + (size_t)N * 128;  // N*4
    float* denom = m     + (size_t)N * 4;    // N*4
    float* a     = denom + (size_t)N * 4;    // E*4 (logits -> ex in place)

    // 1) init segment-max / denom / agg
    init_ws_kernel<<<(N * 128 + 255) / 256, 256, 0, stream>>>(m, denom, agg, N);

    // 2) fused QKVS projections via f32 WMMA (32-row x 16-col tile per wave)
    dim3 gGemm((N + 31) / 32, 4);
    qkvs_gemm_kernel<<<gGemm, 256, 0, stream>>>(feat, Wq, bq, Wk, bk, Wv, bv,
                                                Ws, bs, q, k, v, sk, N);

    // 3) edge logits + segment max (wave per edge; 8 waves per block)
    edge_logits_kernel<<<(E + 7) / 8, 256, 0, stream>>>(q, k, src, dst, a, m, E);

    // 4) exp + denom
    edge_exp_kernel<<<(E * 4 + 255) / 256, 256, 0, stream>>>(a, dst, m, denom, E);

    // 5) weighted aggregation
    edge_aggregate_kernel<<<(E + 7) / 8, 256, 0, stream>>>(v, src, dst, a, denom, agg, E);

    // 6) gate + LayerNorm + PReLU -> output
    gate_ln_kernel<<<(N + 7) / 8, 256, 0, stream>>>(sk, agg, Wg, bg, gamma, beta,
                                                    alpha, (float*)d_out, N);
}